// EMDLoss_19653770346684
// MI455X (gfx1250) — compile-verified
//
#include <hip/hip_runtime.h>
#include <hip/hip_bf16.h>
#include <math.h>

typedef __attribute__((ext_vector_type(2))) float v2f;
typedef __attribute__((ext_vector_type(8))) float v8f;

#define HW      6400
#define NEGV    (-1e9f)
#define THRESHV (1e-3f)

// eps = EPS_FRAC * max(M); max(M) = 2*79^2 = 12482 exactly (integer coords, exact in fp32)
#define EPS_CONST   (12482.0f * 0.01f)
#define LOG2E_F     1.4426950408889634f
#define LN2_F       0.6931471805599453f

// Raw v_exp_f32 / v_log_f32: flush-to-zero on deep underflow is exactly what a
// max-stabilized LSE wants; skips OCML's 5-op denormal range guard.
__device__ __forceinline__ float fast_exp2(float x) { return __builtin_amdgcn_exp2f(x); }
__device__ __forceinline__ float fast_log2(float x) { return __builtin_amdgcn_logf(x); }

// ---------------------------------------------------------------------------
// zero f and g
__global__ void zero_fg(float* __restrict__ f, float* __restrict__ g) {
    int i = blockIdx.x * 256 + threadIdx.x;
    if (i < HW) { f[i] = 0.0f; g[i] = 0.0f; }
}

// ---------------------------------------------------------------------------
// masked log_softmax over 6400 elements + count of active pixels.
// block 0: pred -> log_a, counts[0]=n_pre ; block 1: gt -> log_b, counts[1]=n_gt
__global__ __launch_bounds__(1024) void lsm_setup(const float* __restrict__ pred,
                                                  const float* __restrict__ gt,
                                                  float* __restrict__ log_a,
                                                  float* __restrict__ log_b,
                                                  float* __restrict__ counts) {
    __shared__ float red[1024];
    const float* src = (blockIdx.x == 0) ? pred : gt;
    float*       dst = (blockIdx.x == 0) ? log_a : log_b;
    const int tid = threadIdx.x;

    // pass A: mask + local max
    float lmax = -3.0e38f;
    for (int i = tid; i < HW; i += 1024) {
        float x = src[i];
        float v = (x > THRESHV) ? x : NEGV;
        dst[i] = v;
        lmax = fmaxf(lmax, v);
    }
    red[tid] = lmax; __syncthreads();
    for (int s = 512; s > 0; s >>= 1) {
        if (tid < s) red[tid] = fmaxf(red[tid], red[tid + s]);
        __syncthreads();
    }
    const float gmax = red[0];
    __syncthreads();

    // pass B: sum exp(v - max) and count
    float lsum = 0.0f, lcnt = 0.0f;
    for (int i = tid; i < HW; i += 1024) {
        float v = dst[i];
        lsum += fast_exp2((v - gmax) * LOG2E_F);
        lcnt += (v > THRESHV) ? 1.0f : 0.0f;
    }
    red[tid] = lsum; __syncthreads();
    for (int s = 512; s > 0; s >>= 1) {
        if (tid < s) red[tid] += red[tid + s];
        __syncthreads();
    }
    const float S = red[0];
    __syncthreads();

    red[tid] = lcnt; __syncthreads();
    for (int s = 512; s > 0; s >>= 1) {
        if (tid < s) red[tid] += red[tid + s];
        __syncthreads();
    }
    if (tid == 0) counts[blockIdx.x] = red[0];

    const float lse = gmax + fast_log2(S) * LN2_F;
    for (int i = tid; i < HW; i += 1024) dst[i] = dst[i] - lse;
}

// ---------------------------------------------------------------------------
// One Sinkhorn half step:  out[i] = eps*(logx[i] - LSE_j((v[j] - M_ij)/eps))
// M_ij computed on the fly from pixel coords; M symmetric so same kernel
// serves both directions. Block = 256 (8 waves), handles 16 rows; wave w
// covers 800 columns. Row sums accumulated via V_WMMA_F32_16X16X4_F32 with
// an all-ones B matrix (matrix pipe co-executes with v_exp on trans pipe).
__global__ __launch_bounds__(256) void sinkhorn_half(const float* __restrict__ v,
                                                     const float* __restrict__ logx,
                                                     float* __restrict__ out) {
    __shared__ float pm[8][16];  // per-wave row maxima (log2 domain)
    __shared__ float ps[8][16];  // per-wave row sums of 2^(t - m)

    const float KK = LOG2E_F / EPS_CONST;   // (.)/eps in log2 units
    const int tid  = threadIdx.x;
    const int wave = tid >> 5;
    const int lane = tid & 31;

    const int rowbase = blockIdx.x * 16;    // 16-row tile never crosses a y-line (80%16==0)
    const float fyr = (float)(rowbase / 80);
    const float fxr = (float)((rowbase % 80) + (lane & 15));

    const int laneoff = (lane >> 4) << 1;   // 0 for lanes 0-15, 2 for lanes 16-31 (A-layout K cols)
    const int cstart  = wave * 800 + laneoff;
    const float yc0   = (float)(wave * 10); // 800 cols = 10 pixel-rows
    const float xc0   = (float)laneoff;

    // ---- pass 1: segment row-max of t = (v_j - M_ij)*KK ----
    float m = -3.0e38f;
    {
        const float* p = v + cstart;
        __builtin_prefetch(p, 0, 1);
        float yc = yc0, xc = xc0;
        #pragma unroll 4
        for (int t = 0; t < 200; ++t) {
            float2 gv = *(const float2*)p; p += 4;
            float dy  = fyr - yc;
            float dy2 = dy * dy;
            float dx0 = fxr - xc;
            float dx1 = dx0 - 1.0f;
            float t0  = (gv.x - fmaf(dx0, dx0, dy2)) * KK;
            float t1  = (gv.y - fmaf(dx1, dx1, dy2)) * KK;
            m = fmaxf(m, fmaxf(t0, t1));
            xc += 4.0f;
            if (xc >= 80.0f) { xc -= 80.0f; yc += 1.0f; }
        }
    }
    // lanes l and l+16 cover cols {0,1} / {2,3} mod 4 of the same row -> merge
    m = fmaxf(m, __shfl_xor(m, 16, 32));

    // ---- pass 2: WMMA-accumulated sum of 2^((v_j - M_ij)*KK - m) ----
    // Two accumulators break the WMMA->WMMA RAW chain on the XDL pipe.
    v8f acc0 = {}, acc1 = {};
    v2f bones; bones.x = 1.0f; bones.y = 1.0f;  // all-ones B => D[m,n] = sum_k A[m,k] + C[m,n]
    const float negm = -m;
    {
        const float* p = v + cstart;
        __builtin_prefetch(p, 0, 1);
        float yc = yc0, xc = xc0;
        #pragma unroll 4
        for (int t = 0; t < 200; t += 2) {
            // chunk A -> acc0
            {
                float2 gv = *(const float2*)p;
                float dy  = fyr - yc;
                float dy2 = dy * dy;
                float dx0 = fxr - xc;
                float dx1 = dx0 - 1.0f;
                float d0  = gv.x - fmaf(dx0, dx0, dy2);
                float d1  = gv.y - fmaf(dx1, dx1, dy2);
                v2f a;
                a.x = fast_exp2(fmaf(d0, KK, negm));
                a.y = fast_exp2(fmaf(d1, KK, negm));
                acc0 = __builtin_amdgcn_wmma_f32_16x16x4_f32(
                    false, a, false, bones, (short)0, acc0, false, false);
                xc += 4.0f;
                if (xc >= 80.0f) { xc -= 80.0f; yc += 1.0f; }
            }
            // chunk B -> acc1
            {
                float2 gv = *(const float2*)(p + 4);
                float dy  = fyr - yc;
                float dy2 = dy * dy;
                float dx0 = fxr - xc;
                float dx1 = dx0 - 1.0f;
                float d0  = gv.x - fmaf(dx0, dx0, dy2);
                float d1  = gv.y - fmaf(dx1, dx1, dy2);
                v2f a;
                a.x = fast_exp2(fmaf(d0, KK, negm));
                a.y = fast_exp2(fmaf(d1, KK, negm));
                acc1 = __builtin_amdgcn_wmma_f32_16x16x4_f32(
                    false, a, false, bones, (short)0, acc1, false, false);
                xc += 4.0f;
                if (xc >= 80.0f) { xc -= 80.0f; yc += 1.0f; }
            }
            p += 8;
        }
    }

    // stash per-wave partials (C layout: VGPR r = row r for lanes 0-15, row r+8 for 16-31)
    if ((lane & 15) == 0) {
        const int half = lane >> 4;
        #pragma unroll
        for (int r = 0; r < 8; ++r) ps[wave][half * 8 + r] = acc0[r] + acc1[r];
    }
    if (lane < 16) pm[wave][lane] = m;
    __syncthreads();

    // merge 8 wave-partials per row and finalize
    if (tid < 16) {
        float mm = pm[0][tid];
        #pragma unroll
        for (int w2 = 1; w2 < 8; ++w2) mm = fmaxf(mm, pm[w2][tid]);
        float S = 0.0f;
        #pragma unroll
        for (int w2 = 0; w2 < 8; ++w2) S += ps[w2][tid] * fast_exp2(pm[w2][tid] - mm);
        float lse2 = mm + fast_log2(S);   // log2-domain LSE
        out[rowbase + tid] = EPS_CONST * logx[rowbase + tid]
                           - (EPS_CONST * LN2_F) * lse2;
    }
}

// ---------------------------------------------------------------------------
// per-block partial of sum_ij exp((f_i + g_j - M_ij)/eps) * M_ij
__global__ __launch_bounds__(256) void loss_partial(const float* __restrict__ f,
                                                    const float* __restrict__ g,
                                                    float* __restrict__ partial) {
    __shared__ float fs[16];
    __shared__ float wsum[8];
    const float KK = LOG2E_F / EPS_CONST;
    const int tid  = threadIdx.x;
    const int wave = tid >> 5;
    const int lane = tid & 31;
    const int rowbase = blockIdx.x * 16;

    if (tid < 16) fs[tid] = f[rowbase + tid];
    __syncthreads();

    const float fyr = (float)(rowbase / 80);
    const float fxb = (float)(rowbase % 80);

    float acc = 0.0f;
    const int c0 = wave * 800 + lane;
    for (int it = 0; it < 25; ++it) {
        int c = c0 + it * 32;
        float gv  = g[c];
        float yc  = (float)(c / 80);
        float xc  = (float)(c % 80);
        float dy  = fyr - yc;
        float dy2 = dy * dy;
        #pragma unroll
        for (int r = 0; r < 16; ++r) {
            float dx = (fxb + (float)r) - xc;
            float M  = fmaf(dx, dx, dy2);
            float e  = fast_exp2((fs[r] + gv - M) * KK);
            acc = fmaf(e, M, acc);
        }
    }
    #pragma unroll
    for (int off = 16; off > 0; off >>= 1) acc += __shfl_xor(acc, off, 32);
    if (lane == 0) wsum[wave] = acc;
    __syncthreads();
    if (tid == 0) {
        float s = 0.0f;
        #pragma unroll
        for (int w2 = 0; w2 < 8; ++w2) s += wsum[w2];
        partial[blockIdx.x] = s;
    }
}

// ---------------------------------------------------------------------------
__global__ __launch_bounds__(512) void finalize_loss(const float* __restrict__ partial,
                                                     const float* __restrict__ counts,
                                                     float* __restrict__ outp) {
    __shared__ float red[512];
    const int tid = threadIdx.x;
    float s = 0.0f;
    for (int i = tid; i < 400; i += 512) s += partial[i];
    red[tid] = s; __syncthreads();
    for (int off = 256; off > 0; off >>= 1) {
        if (tid < off) red[tid] += red[tid + off];
        __syncthreads();
    }
    if (tid == 0) outp[0] = red[0] / counts[1] / counts[0];  // /n_gt/n_pre
}

// ---------------------------------------------------------------------------
extern "C" void kernel_launch(void* const* d_in, const int* in_sizes, int n_in,
                              void* d_out, int out_size, void* d_ws, size_t ws_size,
                              hipStream_t stream) {
    const float* pred = (const float*)d_in[0];   // (2,1,80,80) -> uses first 6400
    const float* gt   = (const float*)d_in[1];

    float* ws      = (float*)d_ws;
    float* log_a   = ws;            // 6400
    float* log_b   = ws + 6400;     // 6400
    float* f       = ws + 12800;    // 6400
    float* g       = ws + 19200;    // 6400
    float* counts  = ws + 25600;    // 2   (n_pre, n_gt)
    float* partial = ws + 25604;    // 400

    zero_fg<<<25, 256, 0, stream>>>(f, g);
    lsm_setup<<<2, 1024, 0, stream>>>(pred, gt, log_a, log_b, counts);

    for (int it = 0; it < 100; ++it) {
        sinkhorn_half<<<400, 256, 0, stream>>>(g, log_a, f);  // f-update (rows)
        sinkhorn_half<<<400, 256, 0, stream>>>(f, log_b, g);  // g-update (cols, M symmetric)
    }

    loss_partial<<<400, 256, 0, stream>>>(f, g, partial);
    finalize_loss<<<1, 512, 0, stream>>>(partial, counts, (float*)d_out);
}